// Grid_36283883717059
// MI455X (gfx1250) — compile-verified
//
#include <hip/hip_runtime.h>
#include <math.h>

#define NLVL 16
#define NDENSE 6                  // levels 0..5 dense (res^3 < 2^19), 6..15 hashed — statically certain
#define MAX_ENTRY 524288          // 2^19
#define HMASK (MAX_ENTRY - 1)
#define FDIM 2
#define BLOCK 256

typedef __attribute__((ext_vector_type(2))) float v2f;
typedef __attribute__((ext_vector_type(4))) float v4f;

struct LevelParams {
  float res[NLVL];    // resolution per level (float64-computed, matches numpy)
  float hi[NLVL];     // f32(res - 1.0001) clamp bound
};

__global__ __launch_bounds__(BLOCK) void ngp_hashgrid_kernel(
    const float* __restrict__ xin,
    const float* __restrict__ tables,
    float* __restrict__ out,
    int n_points, LevelParams lp)
{
  __shared__ float sx[BLOCK * 3];
  const int tid = threadIdx.x;
  int n = blockIdx.x * BLOCK + tid;
  if (n >= n_points) n = n_points - 1;  // tail lanes duplicate last point (benign, deterministic)

  // ---- CDNA5 async global->LDS staging of this thread's (x,y,z) coords ----
  // inst offset applies to both LDS destination and global address; ASYNCcnt
  // tracks completion (s_wait_asynccnt).
  {
    unsigned lds  = (unsigned)(size_t)(&sx[tid * 3]);  // low 32 bits = LDS byte offset
    unsigned goff = (unsigned)n * 12u;
    asm volatile(
        "global_load_async_to_lds_b32 %0, %1, %2 offset:0\n\t"
        "global_load_async_to_lds_b32 %0, %1, %2 offset:4\n\t"
        "global_load_async_to_lds_b32 %0, %1, %2 offset:8\n\t"
        "s_wait_asynccnt 0x0"
        :: "v"(lds), "v"(goff), "s"(xin)
        : "memory");
  }
  const float px = sx[tid*3+0];
  const float py = sx[tid*3+1];
  const float pz = sx[tid*3+2];

  float acc[2*NLVL];

  #pragma unroll
  for (int l = 0; l < NLVL; ++l) {
    const float res = lp.res[l];
    const float hiv = lp.hi[l];
    const float rm1 = res - 1.0f;

    // coord = clip(x*(res-1), 0, res-1.0001); fmaxf/fminf also sanitize any
    // garbage (NaN) from duplicated tail lanes so indices stay in-bounds.
    float cx = fminf(fmaxf(px * rm1, 0.0f), hiv);
    float cy = fminf(fmaxf(py * rm1, 0.0f), hiv);
    float cz = fminf(fmaxf(pz * rm1, 0.0f), hiv);
    float fx = floorf(cx), fy = floorf(cy), fz = floorf(cz);
    float dx = cx - fx, dy = cy - fy, dz = cz - fz;
    unsigned ix = (unsigned)(int)fx, iy = (unsigned)(int)fy, iz = (unsigned)(int)fz;

    unsigned idx[8];
    if (l < NDENSE) {             // compile-time constant under full unroll
      // dense level: x + y*res + z*res^2 (all values < 2^24, exact)
      unsigned r = (unsigned)res, r2 = r * r;
      unsigned b = ix + iy * r + iz * r2;
      idx[0]=b;     idx[1]=b+r2;     idx[2]=b+r;     idx[3]=b+r+r2;
      idx[4]=b+1;   idx[5]=b+1+r2;   idx[6]=b+1+r;   idx[7]=b+1+r+r2;
    } else {
      // hash level: (x*P0 ^ y*P1 ^ z*P2) mod 2^19 — only low bits matter, so
      // u32 multiplies reproduce the reference's int64 math bit-exactly.
      const unsigned P0 = 3367900313u, P1 = 2654435761u, P2 = 805459861u;
      unsigned hx0 = ix * P0, hx1 = hx0 + P0;
      unsigned hy0 = iy * P1, hy1 = hy0 + P1;
      unsigned hz0 = iz * P2, hz1 = hz0 + P2;
      idx[0]=(hx0^hy0^hz0)&HMASK; idx[1]=(hx0^hy0^hz1)&HMASK;
      idx[2]=(hx0^hy1^hz0)&HMASK; idx[3]=(hx0^hy1^hz1)&HMASK;
      idx[4]=(hx1^hy0^hz0)&HMASK; idx[5]=(hx1^hy0^hz1)&HMASK;
      idx[6]=(hx1^hy1^hz0)&HMASK; idx[7]=(hx1^hy1^hz1)&HMASK;
    }

    // 8 independent 8-byte gathers (global_load_b64), issued as a group so
    // LOADcnt stays deep; tables (64 MB total) are L2-resident (192 MB L2).
    const v2f* __restrict__ tb =
        (const v2f*)(tables + (size_t)l * (size_t)(MAX_ENTRY * FDIM));
    v2f f0=tb[idx[0]], f1=tb[idx[1]], f2=tb[idx[2]], f3=tb[idx[3]];
    v2f f4=tb[idx[4]], f5=tb[idx[5]], f6=tb[idx[6]], f7=tb[idx[7]];

    float mx=1.f-dx, my=1.f-dy, mz=1.f-dz;
    float w0=mx*my*mz, w1=mx*my*dz, w2=mx*dy*mz, w3=mx*dy*dz;
    float w4=dx*my*mz, w5=dx*my*dz, w6=dx*dy*mz, w7=dx*dy*dz;

    float a0 = w0*f0.x + w1*f1.x + w2*f2.x + w3*f3.x
             + w4*f4.x + w5*f5.x + w6*f6.x + w7*f7.x;
    float a1 = w0*f0.y + w1*f1.y + w2*f2.y + w3*f3.y
             + w4*f4.y + w5*f5.y + w6*f6.y + w7*f7.y;
    acc[2*l]   = a0;
    acc[2*l+1] = a1;
  }

  // ---- streaming output: 8x b128 non-temporal stores (th:TH_STORE_NT).
  // The 256 MB write stream must NOT evict the L2-resident tables.
  v4f* o = (v4f*)(out + (size_t)n * (size_t)(2*NLVL));
  #pragma unroll
  for (int i = 0; i < 8; ++i) {
    v4f v = { acc[4*i+0], acc[4*i+1], acc[4*i+2], acc[4*i+3] };
    __builtin_nontemporal_store(v, o + i);
  }
}

extern "C" void kernel_launch(void* const* d_in, const int* in_sizes, int n_in,
                              void* d_out, int out_size, void* d_ws, size_t ws_size,
                              hipStream_t stream) {
  const float* x      = (const float*)d_in[0];   // (N,3) f32
  const float* tables = (const float*)d_in[1];   // (16, 2^19, 2) f32
  float* out = (float*)d_out;                    // (N,32) f32
  const int n_points = in_sizes[0] / 3;

  // Reproduce numpy's float64 level setup exactly (same formula, float64 ops),
  // so boundary cases (e.g. level 15: 16*b^15 ≈ 2048 ± eps) match the reference.
  LevelParams lp;
  const double b = exp((log(2048.0) - log(16.0)) / 15.0);
  for (int i = 0; i < NLVL; ++i) {
    double r = floor(16.0 * pow(b, (double)i));
    lp.res[i] = (float)r;
    lp.hi[i]  = (float)(r - 1.0001);
  }

  const int blocks = (n_points + BLOCK - 1) / BLOCK;
  ngp_hashgrid_kernel<<<blocks, BLOCK, 0, stream>>>(x, tables, out, n_points, lp);
}